// DynamicConv1d_82867099009254
// MI455X (gfx1250) — compile-verified
//
#include <hip/hip_runtime.h>
#include <cmath>

// Problem constants (from reference)
#define DIMC 512      // C
#define GK   3        // K (conv taps)
#define GG   4        // G (groups)
#define BB   64       // B
#define LL   1000     // L
#define HID  128      // C // R
#define OC   2048     // C * G
#define NCOL 256      // B * 4 columns (3 pool bins + mean per batch)

typedef __attribute__((ext_vector_type(2))) float v2f;
typedef __attribute__((ext_vector_type(8))) float v8f;

// ---------------------------------------------------------------------------
// Kernel 1: adaptive pool (3 overlapping bins) + full mean per (b, c).
// x is [B, L, C]; one block per b, one thread per channel -> fully coalesced.
// Output Y is [C rows, NCOL cols] row-major: Y[c*NCOL + b*4 + t], t=0..2 pool
// bins, t=3 full mean.  This is the K-side operand of GEMM1.
// ---------------------------------------------------------------------------
__global__ void pool_kernel(const float* __restrict__ x, float* __restrict__ Y) {
  const int b = blockIdx.x;
  const int c = threadIdx.x;                 // 0..511
  const float* xb = x + (size_t)b * LL * DIMC + c;
  constexpr int E0 = (1 * LL + GK - 1) / GK; // 334
  constexpr int S1 = (1 * LL) / GK;          // 333
  constexpr int E1 = (2 * LL + GK - 1) / GK; // 667
  constexpr int S2 = (2 * LL) / GK;          // 666
  float s0 = 0.f, s1 = 0.f, s2 = 0.f, st = 0.f;
  for (int l = 0; l < LL; ++l) {
    float v = xb[(size_t)l * DIMC];
    st += v;
    if (l < E0)            s0 += v;
    if (l >= S1 && l < E1) s1 += v;
    if (l >= S2)           s2 += v;
  }
  float* yr = Y + (size_t)c * NCOL + b * 4;
  yr[0] = s0 / (float)E0;
  yr[1] = s1 / (float)(E1 - S1);
  yr[2] = s2 / (float)(LL - S2);
  yr[3] = st / (float)LL;
}

// ---------------------------------------------------------------------------
// WMMA f32 GEMM: Out[M, NCOL] = act(W[M, KD] * Yb[KD, NCOL] + bias).
// One wave per 16x16 output tile, V_WMMA_F32_16X16X4_F32 over the K dim.
// Operand layouts per CDNA5 ISA 7.12.2:
//   A 16x4 : lanes 0-15 hold M=lane (K=0,1 in vgpr 0,1); lanes 16-31 K=2,3
//   B 4x16 : lanes 0-15 hold N=lane (K=0,1);            lanes 16-31 K=2,3
//   D 16x16: vgpr v -> row v + 8*(lane>=16), col = lane%16
// ---------------------------------------------------------------------------
template <int KD, bool DO_GELU>
__global__ void gemm_wmma(const float* __restrict__ W,
                          const float* __restrict__ Yb,
                          const float* __restrict__ bias,
                          float* __restrict__ Out) {
  const int row0 = blockIdx.x * 16;
  const int col0 = blockIdx.y * 16;
  const int lane = threadIdx.x;       // 0..31, one wave per block
  const int half = lane >> 4;         // 0: K pair {0,1}, 1: K pair {2,3}
  const int lid  = lane & 15;         // M (for A) / N (for B)

  const float* wr = W  + (size_t)(row0 + lid) * KD + half * 2;
  const float* br = Yb + (size_t)(half * 2) * NCOL + col0 + lid;

  v8f acc = {};
#pragma unroll 4
  for (int kk = 0; kk < KD; kk += 4) {
    v2f a = *(const v2f*)(wr + kk);   // two consecutive K elems, 8B aligned
    v2f bv;
    bv.x = br[(size_t)kk * NCOL];
    bv.y = br[(size_t)kk * NCOL + NCOL];
    acc = __builtin_amdgcn_wmma_f32_16x16x4_f32(
        /*neg_a=*/false, a, /*neg_b=*/false, bv,
        /*c_mod=*/(short)0, acc, /*reuse_a=*/false, /*reuse_b=*/false);
  }

#pragma unroll
  for (int v = 0; v < 8; ++v) {
    const int r = row0 + v + half * 8;
    float val = acc[v] + bias[r];
    if (DO_GELU) {
      // exact GELU: 0.5*x*(1+erf(x/sqrt(2)))
      val = 0.5f * val * (1.0f + erff(val * 0.70710678118654752f));
    }
    Out[(size_t)r * NCOL + col0 + lid] = val;
  }
}

// ---------------------------------------------------------------------------
// Kernel 4: softmax over groups + expert mixing.
// S is [OC rows = g*C + c, NCOL]; produce WD[b][4][C] = {tap0,tap1,tap2,bias}.
// ---------------------------------------------------------------------------
__global__ void mix_kernel(const float* __restrict__ S,
                           const float* __restrict__ weight,   // [G, C, K]
                           const float* __restrict__ bias_p,   // [G, C]
                           float* __restrict__ WD) {            // [B, 4, C]
  const int idx = blockIdx.x * blockDim.x + threadIdx.x;
  if (idx >= BB * DIMC) return;
  const int b = idx / DIMC;
  const int c = idx % DIMC;

  float sv[GG][4];
#pragma unroll
  for (int g = 0; g < GG; ++g)
#pragma unroll
    for (int t = 0; t < 4; ++t)
      sv[g][t] = S[(size_t)(g * DIMC + c) * NCOL + b * 4 + t];

  float p[GG][4];
#pragma unroll
  for (int t = 0; t < 4; ++t) {
    float m = sv[0][t];
#pragma unroll
    for (int g = 1; g < GG; ++g) m = fmaxf(m, sv[g][t]);
    float sum = 0.f;
#pragma unroll
    for (int g = 0; g < GG; ++g) { p[g][t] = expf(sv[g][t] - m); sum += p[g][t]; }
    const float inv = 1.0f / sum;
#pragma unroll
    for (int g = 0; g < GG; ++g) p[g][t] *= inv;
  }

  float* wd = WD + (size_t)b * 4 * DIMC;
#pragma unroll
  for (int k = 0; k < GK; ++k) {
    float acc = 0.f;
#pragma unroll
    for (int g = 0; g < GG; ++g)
      acc += p[g][k] * weight[(size_t)(g * DIMC + c) * GK + k];
    wd[k * DIMC + c] = acc;
  }
  float bd = 0.f;
#pragma unroll
  for (int g = 0; g < GG; ++g) bd += p[g][3] * bias_p[g * DIMC + c];
  wd[3 * DIMC + c] = bd;
}

// ---------------------------------------------------------------------------
// Kernel 5: depthwise 'same' K=3 conv with per-(b,c) taps, streaming float4.
// x/out are [B, L, C]; each thread owns 4 channels, sliding-window over L.
// ---------------------------------------------------------------------------
#define LCHUNK 100
__global__ void conv_kernel(const float* __restrict__ x,
                            const float* __restrict__ WD,
                            float* __restrict__ out) {
  const int b   = blockIdx.y;
  const int l0  = blockIdx.x * LCHUNK;
  const int tid = threadIdx.x;                       // 0..127 -> 4 channels each
  const float4* Xb = (const float4*)(x + (size_t)b * LL * DIMC);
  float4*       Ob = (float4*)(out + (size_t)b * LL * DIMC);
  const float* wdb = WD + (size_t)b * 4 * DIMC;
  const float4 w0 = ((const float4*)(wdb + 0 * DIMC))[tid];
  const float4 w1 = ((const float4*)(wdb + 1 * DIMC))[tid];
  const float4 w2 = ((const float4*)(wdb + 2 * DIMC))[tid];
  const float4 bd = ((const float4*)(wdb + 3 * DIMC))[tid];
  const float4 zero = make_float4(0.f, 0.f, 0.f, 0.f);

  constexpr int C4 = DIMC / 4;                       // 128 float4 per row
  float4 xm1 = (l0 > 0) ? Xb[(size_t)(l0 - 1) * C4 + tid] : zero;
  float4 xc  = Xb[(size_t)l0 * C4 + tid];
  for (int l = l0; l < l0 + LCHUNK; ++l) {
    float4 xp1 = (l + 1 < LL) ? Xb[(size_t)(l + 1) * C4 + tid] : zero;
    if (l + 8 < LL)                                  // global_prefetch_b8
      __builtin_prefetch(&Xb[(size_t)(l + 8) * C4 + tid], 0, 0);
    float4 r;
    r.x = fmaf(w0.x, xm1.x, fmaf(w1.x, xc.x, fmaf(w2.x, xp1.x, bd.x)));
    r.y = fmaf(w0.y, xm1.y, fmaf(w1.y, xc.y, fmaf(w2.y, xp1.y, bd.y)));
    r.z = fmaf(w0.z, xm1.z, fmaf(w1.z, xc.z, fmaf(w2.z, xp1.z, bd.z)));
    r.w = fmaf(w0.w, xm1.w, fmaf(w1.w, xc.w, fmaf(w2.w, xp1.w, bd.w)));
    Ob[(size_t)l * C4 + tid] = r;
    xm1 = xc; xc = xp1;
  }
}

// ---------------------------------------------------------------------------
extern "C" void kernel_launch(void* const* d_in, const int* in_sizes, int n_in,
                              void* d_out, int out_size, void* d_ws, size_t ws_size,
                              hipStream_t stream) {
  const float* x      = (const float*)d_in[0];  // [64, 1000, 512]
  const float* weight = (const float*)d_in[1];  // [4, 512, 3]
  const float* bias_p = (const float*)d_in[2];  // [4, 512]
  const float* w1     = (const float*)d_in[3];  // [128, 512]
  const float* b1     = (const float*)d_in[4];  // [128]
  const float* w2     = (const float*)d_in[5];  // [2048, 128]
  const float* b2     = (const float*)d_in[6];  // [2048]
  float* out = (float*)d_out;                   // [64, 1000, 512]

  float* Y  = (float*)d_ws;                     // [512, 256]
  float* H  = Y  + (size_t)DIMC * NCOL;         // [128, 256]
  float* S  = H  + (size_t)HID  * NCOL;         // [2048, 256]
  float* WD = S  + (size_t)OC   * NCOL;         // [64, 4, 512]

  // 1) pooled bins + mean  -> Y [512 x 256]
  pool_kernel<<<BB, DIMC, 0, stream>>>(x, Y);

  // 2) H = GELU(w1 @ Y + b1)  [128 x 256], WMMA f32 16x16x4
  gemm_wmma<DIMC, true><<<dim3(HID / 16, NCOL / 16), 32, 0, stream>>>(w1, Y, b1, H);

  // 3) S = w2 @ H + b2  [2048 x 256], WMMA f32 16x16x4
  gemm_wmma<HID, false><<<dim3(OC / 16, NCOL / 16), 32, 0, stream>>>(w2, H, b2, S);

  // 4) softmax over groups, mix expert taps + bias -> WD [64 x 4 x 512]
  mix_kernel<<<(BB * DIMC + 255) / 256, 256, 0, stream>>>(S, weight, bias_p, WD);

  // 5) depthwise conv, streaming
  conv_kernel<<<dim3(LL / LCHUNK, BB), DIMC / 4, 0, stream>>>(x, WD, out);
}